// RWaveCenteredPatchEmbedding_44023414784128
// MI455X (gfx1250) — compile-verified
//
#include <hip/hip_runtime.h>

typedef __attribute__((ext_vector_type(2))) float v2f;
typedef __attribute__((ext_vector_type(8))) float v8f;

#define D_MODEL   128
#define PATCH_LEN 64
#define STRIDE_   32
#define HALF_     32
#define POS_DIM   32
#define KEEP_     96
#define BATCH_    128
#define SEQ_LEN_  5000
#define N_VARS_   12
#define N_PEAKS_  16
#define M_CENT    155              // (5000-32)/32
#define N_PATCH   (N_PEAKS_*M_CENT) // 2480
#define TILES_SEG 10               // ceil(155/16)
#define LDSW      66               // 64 + 2 pad (bank-conflict-free B reads)

// ---------------- Kernel 1: channel mean xm[b,s] = mean_v x[b,s,v] ----------
__global__ __launch_bounds__(256)
void xm_mean_kernel(const float* __restrict__ x, float* __restrict__ xm, int total) {
    int i = blockIdx.x * blockDim.x + threadIdx.x;
    if (i >= total) return;
    // 12 floats = 48 bytes, always 16B aligned -> three float4 loads
    const float4* p4 = (const float4*)(x + (size_t)i * N_VARS_);
    float4 a = p4[0], b = p4[1], c = p4[2];
    float s = a.x + a.y + a.z + a.w + b.x + b.y + b.z + b.w + c.x + c.y + c.z + c.w;
    xm[i] = s * (1.0f / 12.0f);
}

// ---------------- Kernel 2: fused patch-GEMM (WMMA f32) + phase embedding ---
__global__ __launch_bounds__(128)
void rwave_embed_kernel(const float* __restrict__ xm,
                        const float* __restrict__ Wv,   // [128,64]
                        const float* __restrict__ bv,   // [128]
                        const float* __restrict__ Wp,   // [32,2]
                        const float* __restrict__ bp,   // [32]
                        const int*   __restrict__ peaks,// [B,16]
                        float* __restrict__ out) {
    __shared__ float WvLds[D_MODEL * LDSW];

    const int tid = threadIdx.x;
    // Stage Wv into LDS with padded rows: WvLds[d*66 + p]
    for (int i = tid; i < D_MODEL * PATCH_LEN; i += 128) {
        int d = i >> 6, p = i & 63;
        WvLds[d * LDSW + p] = Wv[i];
    }
    __syncthreads();

    const int wave = tid >> 5;
    const int lane = tid & 31;
    const int lm = lane & 15;   // A: row M ; B/D: column N
    const int lh = lane >> 4;   // K-pair half selector

    // tile id -> (batch, peak, row-tile)
    int t    = blockIdx.x * 4 + wave;                // < 128*16*10 = 20480
    int b    = t / (N_PEAKS_ * TILES_SEG);
    int rem  = t - b * (N_PEAKS_ * TILES_SEG);
    int kpk  = rem / TILES_SEG;
    int tile = rem - kpk * TILES_SEG;
    int m0   = tile * 16;

    const int* pk = peaks + b * N_PEAKS_;
    const int mypk = pk[kpk];
    const int res  = ((mypk % STRIDE_) + STRIDE_) % STRIDE_;

    // ---- Load A fragments (16 rows x K=64, fp32) --------------------------
    int m_un = m0 + lm;                              // unclamped row (validity)
    int m_cl = m_un > (M_CENT - 1) ? (M_CENT - 1) : m_un;
    int center = res + STRIDE_ * (m_cl + 1);
    const float* arow = xm + (size_t)b * SEQ_LEN_ + (center - HALF_);
    v2f a[16];
#pragma unroll
    for (int ks = 0; ks < 16; ++ks) {
        int kk = 4 * ks + 2 * lh;
        a[ks].x = arow[kk];
        a[ks].y = arow[kk + 1];
    }

    const size_t rowbase = (size_t)b * N_PATCH + (size_t)kpk * M_CENT;

    // ---- 6 column tiles (d = 0..95), K=64 via 16 x WMMA_F32_16X16X4_F32 ---
#pragma unroll
    for (int ct = 0; ct < 6; ++ct) {
        v8f acc = {0.f, 0.f, 0.f, 0.f, 0.f, 0.f, 0.f, 0.f};
        const float* wrow = &WvLds[(16 * ct + lm) * LDSW];
#pragma unroll
        for (int ks = 0; ks < 16; ++ks) {
            int kk = 4 * ks + 2 * lh;
            v2f bf;
            bf.x = wrow[kk];
            bf.y = wrow[kk + 1];
            acc = __builtin_amdgcn_wmma_f32_16x16x4_f32(
                false, a[ks], false, bf, (short)0, acc, false, false);
        }
        float bias = bv[16 * ct + lm];               // column-only bias
#pragma unroll
        for (int r = 0; r < 8; ++r) {
            int m = m0 + r + 8 * lh;                 // D row = vgpr + 8*half
            if (m < M_CENT) {
                size_t row = rowbase + (size_t)m;
                out[row * D_MODEL + 16 * ct + lm] = acc[r] + bias;
            }
        }
    }

    // ---- Cardiac-phase positional embedding (d = 96..127) -----------------
    int best = 0, bestd = abs(center - pk[0]);
#pragma unroll
    for (int j = 1; j < N_PEAKS_; ++j) {
        int dd = abs(center - pk[j]);
        if (dd < bestd) { bestd = dd; best = j; }   // first-min like argmin
    }
    int nearest = pk[best];
    int rr;
    if (best + 1 < N_PEAKS_)      rr = pk[best + 1] - nearest;
    else if (best > 0)            rr = nearest - pk[best - 1];
    else                          rr = 128;
    if (rr < 1) rr = 1;
    float rel = (float)(center - nearest) / (float)rr;
    float ph  = rel + 1.0f;
    ph -= floorf(ph);                                // floored mod 1.0
    const float TWO_PI = 6.28318530717958647692f;
    float sv = sinf(TWO_PI * ph);
    float cv = cosf(TWO_PI * ph);

    if (m_un < M_CENT) {
        size_t row = rowbase + (size_t)m_un;
        float pv[16];
#pragma unroll
        for (int j = 0; j < 16; ++j) {
            int q = 16 * lh + j;                     // lane halves split 32 dims
            pv[j] = sv * Wp[2 * q] + cv * Wp[2 * q + 1] + bp[q];
        }
        float* orow = out + row * D_MODEL + KEEP_ + 16 * lh;
#pragma unroll
        for (int j4 = 0; j4 < 4; ++j4) {
            float4 v; v.x = pv[4*j4]; v.y = pv[4*j4+1]; v.z = pv[4*j4+2]; v.w = pv[4*j4+3];
            *(float4*)(orow + 4 * j4) = v;
        }
        if (lh == 0) {
            const size_t pbase = (size_t)BATCH_ * N_PATCH * D_MODEL;
            out[pbase + row]                            = (float)center;
            out[pbase + (size_t)BATCH_ * N_PATCH + row] = (float)PATCH_LEN;
        }
    }
}

extern "C" void kernel_launch(void* const* d_in, const int* in_sizes, int n_in,
                              void* d_out, int out_size, void* d_ws, size_t ws_size,
                              hipStream_t stream) {
    const float* x     = (const float*)d_in[0];   // [128,5000,12]
    const float* Wv    = (const float*)d_in[1];   // [128,64]
    const float* bv    = (const float*)d_in[2];   // [128]
    const float* Wp    = (const float*)d_in[3];   // [32,2]
    const float* bp    = (const float*)d_in[4];   // [32]
    const int*   peaks = (const int*)d_in[5];     // [128,16]
    float* out = (float*)d_out;
    float* xm  = (float*)d_ws;                    // [128,5000] fp32 scratch

    const int total = BATCH_ * SEQ_LEN_;          // 640,000
    xm_mean_kernel<<<(total + 255) / 256, 256, 0, stream>>>(x, xm, total);

    const int nTiles  = BATCH_ * N_PEAKS_ * TILES_SEG;  // 20,480 wave-tiles
    const int nBlocks = nTiles / 4;                     // 4 waves / block
    rwave_embed_kernel<<<nBlocks, 128, 0, stream>>>(xm, Wv, bv, Wp, bp, peaks, out);
}